// grsce_59167469470096
// MI455X (gfx1250) — compile-verified
//
#include <hip/hip_runtime.h>
#include <hip/hip_bf16.h>

// ---- structural constants from the reference ----
#define NN   512000      // total nodes
#define EE   16384000    // total edges
#define H1   128
#define H2   5
#define H2P  8           // padded row stride for msg/agg (16B-aligned float4 + float)
#define GG   512         // graphs
#define NPG  1000        // nodes per graph
#define BB   32          // batch
#define TT   16          // seq len

typedef __attribute__((ext_vector_type(2))) float v2f;
typedef __attribute__((ext_vector_type(8))) float v8f;

// workspace layout (floats)
#define WS_OUTDEG  ((size_t)0)
#define WS_INDEG   ((size_t)NN)
#define WS_MSG     ((size_t)(2*NN))          // [NN][H2P]
#define WS_AGG     ((size_t)(10*NN))         // [NN][H2P]
#define WS_POOLED  ((size_t)(18*NN))         // [GG][H2]

// -------- kernel 1: zero scratch (deg tables + msg + agg) --------
__global__ void zero_ws_kernel(float* __restrict__ ws, size_t n) {
    size_t i = (size_t)blockIdx.x * blockDim.x + threadIdx.x;
    size_t stride = (size_t)gridDim.x * blockDim.x;
    for (; i < n; i += stride) ws[i] = 0.0f;
}

// -------- kernel 2: out/in degree via fp32 hardware atomics --------
__global__ void degree_kernel(const int* __restrict__ src, const int* __restrict__ dst,
                              float* __restrict__ outdeg, float* __restrict__ indeg) {
    unsigned e = blockIdx.x * blockDim.x + threadIdx.x;
    if (e >= EE) return;
    unsafeAtomicAdd(&outdeg[src[e]], 1.0f);
    unsafeAtomicAdd(&indeg[dst[e]], 1.0f);
}

// -------- kernel 3: msg = (feat @ W) * src_norm  via V_WMMA_F32_16X16X4_F32 --------
// One wave32 per 16-node tile. B = W zero-padded+transposed, staged once in LDS so the
// inner loop is branch-free: 1x global_load_b64 (A) + 1x ds_load_b64 (B) + 1x wmma.
// A-frag (ISA 7.12.2, 32-bit A 16x4): lane m=l&15; VGPR0/1 hold K = 2*(l>>4), 2*(l>>4)+1.
// B-frag mirrors: VGPR0/1 hold rows K = 2*(l>>4), +1 at column n=l&15.
// C/D (16x16 f32): VGPR r holds (M = r + 8*(l>>4), N = l&15).
__global__ void xw_wmma_kernel(const float* __restrict__ feat, const float* __restrict__ W,
                               const float* __restrict__ outdeg, float* __restrict__ msg) {
    __shared__ float Wt[16 * H1];            // Wt[n][k] = (n<5) ? W[k][n] : 0   (8 KB)
    for (int idx = threadIdx.x; idx < 16 * H1; idx += blockDim.x) {
        const int n = idx >> 7;              // 0..15
        const int k = idx & (H1 - 1);        // 0..127
        Wt[idx] = (n < H2) ? W[(size_t)k * H2 + n] : 0.0f;
    }
    __syncthreads();

    const int wave = (blockIdx.x * blockDim.x + threadIdx.x) >> 5;
    const int lane = threadIdx.x & 31;
    const int base = wave * 16;
    if (base >= NN) return;                  // uniform per-wave: EXEC stays all-ones
    const int m     = lane & 15;
    const int khalf = lane >> 4;             // 0: K pair {k,k+1}, 1: pair {k+2,k+3}
    const float* frow = feat + (size_t)(base + m) * H1;
    const float* brow = Wt + (m << 7);       // this lane's B column, padded

    v8f c = {0.f,0.f,0.f,0.f,0.f,0.f,0.f,0.f};
    #pragma unroll
    for (int kk = 0; kk < H1; kk += 4) {
        const int ka = kk + 2 * khalf;
        const v2f a  = *(const v2f*)(frow + ka);   // global_load_b64
        const v2f bf = *(const v2f*)(brow + ka);   // ds_load_b64, branch-free
        c = __builtin_amdgcn_wmma_f32_16x16x4_f32(
                /*neg_a=*/false, a, /*neg_b=*/false, bf,
                /*c_mod=*/(short)0, c, /*reuse_a=*/false, /*reuse_b=*/false);
    }
    // write the 5 valid columns (stride-8 padded rows), fused with src_norm
    if (m < H2) {
        #pragma unroll
        for (int r = 0; r < 8; ++r) {
            const int node = base + r + 8 * khalf;
            const float sn = __frsqrt_rn(fmaxf(outdeg[node], 1.0f));
            msg[(size_t)node * H2P + m] = c[r] * sn;
        }
    }
}

// -------- kernel 4: agg[dst] += msg[src]  (b128+b32 gather, 5 hw fp atomics) --------
__global__ void edge_scatter_kernel(const int* __restrict__ src, const int* __restrict__ dst,
                                    const float* __restrict__ msg, float* __restrict__ agg) {
    unsigned e = blockIdx.x * blockDim.x + threadIdx.x;
    if (e >= EE) return;
    const int s = src[e], d = dst[e];
    const float4 v4 = *(const float4*)(msg + (size_t)s * H2P);   // cols 0..3, 16B aligned
    const float  v1 = msg[(size_t)s * H2P + 4];                  // col 4
    float* arow = agg + (size_t)d * H2P;
    unsafeAtomicAdd(&arow[0], v4.x);
    unsafeAtomicAdd(&arow[1], v4.y);
    unsafeAtomicAdd(&arow[2], v4.z);
    unsafeAtomicAdd(&arow[3], v4.w);
    unsafeAtomicAdd(&arow[4], v1);
}

// -------- kernel 5: h = agg*dst_norm + b, then per-graph mean (block g owns graph g) ----
__global__ void pool_kernel(const float* __restrict__ agg, const float* __restrict__ indeg,
                            const float* __restrict__ bvec, float* __restrict__ pooled) {
    __shared__ float sred[256 * H2];
    const int g = blockIdx.x;
    const int tid = threadIdx.x;
    float acc[H2] = {0.f, 0.f, 0.f, 0.f, 0.f};
    for (int i = tid; i < NPG; i += blockDim.x) {
        const int node = g * NPG + i;
        const float dn = __frsqrt_rn(fmaxf(indeg[node], 1.0f));
        const float4 a4 = *(const float4*)(agg + (size_t)node * H2P);
        const float  a1 = agg[(size_t)node * H2P + 4];
        acc[0] += a4.x * dn + bvec[0];
        acc[1] += a4.y * dn + bvec[1];
        acc[2] += a4.z * dn + bvec[2];
        acc[3] += a4.w * dn + bvec[3];
        acc[4] += a1   * dn + bvec[4];
    }
    #pragma unroll
    for (int j = 0; j < H2; ++j) sred[tid * H2 + j] = acc[j];
    __syncthreads();
    for (int s = 128; s > 0; s >>= 1) {
        if (tid < s) {
            #pragma unroll
            for (int j = 0; j < H2; ++j)
                sred[tid * H2 + j] += sred[(tid + s) * H2 + j];
        }
        __syncthreads();
    }
    if (tid == 0) {
        #pragma unroll
        for (int j = 0; j < H2; ++j)
            pooled[(size_t)g * H2 + j] = sred[j] * (1.0f / NPG);
    }
}

// -------- kernel 6: single-wave LSTM (lane = batch elem) + MSE loss reduce --------
__device__ __forceinline__ float sigmoidf_(float x) { return 1.0f / (1.0f + __expf(-x)); }

__global__ void lstm_loss_kernel(const float* __restrict__ pooled,
                                 const float* __restrict__ Wih, const float* __restrict__ Whh,
                                 const float* __restrict__ bih, const float* __restrict__ bhh,
                                 const float* __restrict__ w_r, const float* __restrict__ b_r,
                                 const float* __restrict__ target, float* __restrict__ out) {
    const int lane = threadIdx.x;        // 0..31 == batch index
    float h[H2]  = {0.f, 0.f, 0.f, 0.f, 0.f};
    float cs[H2] = {0.f, 0.f, 0.f, 0.f, 0.f};
    for (int t = 0; t < TT; ++t) {
        float x[H2];
        #pragma unroll
        for (int j = 0; j < H2; ++j)
            x[j] = pooled[(size_t)(lane * TT + t) * H2 + j];
        float gates[4 * H2];
        #pragma unroll
        for (int gi = 0; gi < 4 * H2; ++gi) {
            float a = bih[gi] + bhh[gi];
            #pragma unroll
            for (int j = 0; j < H2; ++j)
                a += Wih[gi * H2 + j] * x[j] + Whh[gi * H2 + j] * h[j];
            gates[gi] = a;
        }
        #pragma unroll
        for (int j = 0; j < H2; ++j) {
            const float ig = sigmoidf_(gates[j]);
            const float fg = sigmoidf_(gates[H2 + j]);
            const float gg = tanhf(gates[2 * H2 + j]);
            const float og = sigmoidf_(gates[3 * H2 + j]);
            cs[j] = fg * cs[j] + ig * gg;
            h[j]  = og * tanhf(cs[j]);
        }
    }
    float pred = b_r[0];
    #pragma unroll
    for (int j = 0; j < H2; ++j) pred += h[j] * w_r[j];
    const float err = pred - target[lane];
    float l = err * err * (1.0f / BB);
    for (int o = 16; o > 0; o >>= 1) l += __shfl_xor(l, o, 32);
    if (lane == 0) out[0] = l;
}

extern "C" void kernel_launch(void* const* d_in, const int* in_sizes, int n_in,
                              void* d_out, int out_size, void* d_ws, size_t ws_size,
                              hipStream_t stream) {
    const float* feat   = (const float*)d_in[0];
    const float* W      = (const float*)d_in[1];
    const float* bvec   = (const float*)d_in[2];
    const float* Wih    = (const float*)d_in[3];
    const float* Whh    = (const float*)d_in[4];
    const float* bih    = (const float*)d_in[5];
    const float* bhh    = (const float*)d_in[6];
    const float* w_r    = (const float*)d_in[7];
    const float* b_r    = (const float*)d_in[8];
    const float* target = (const float*)d_in[9];
    const int*   src    = (const int*)d_in[10];
    const int*   dst    = (const int*)d_in[11];
    // d_in[12] = graph_ids: contiguous repeat(arange(G), NPG) -> implicit in pool_kernel

    float* ws      = (float*)d_ws;
    float* outdeg  = ws + WS_OUTDEG;
    float* indeg   = ws + WS_INDEG;
    float* msg     = ws + WS_MSG;
    float* agg     = ws + WS_AGG;
    float* pooled  = ws + WS_POOLED;

    // 1) zero deg tables + msg + agg (contiguous [0, 18N))
    zero_ws_kernel<<<9000, 256, 0, stream>>>(ws, (size_t)18 * NN);
    // 2) degrees
    degree_kernel<<<EE / 256, 256, 0, stream>>>(src, dst, outdeg, indeg);
    // 3) msg = (feat @ W) * src_norm via f32 WMMA; 16 nodes per wave, 8 waves per block
    xw_wmma_kernel<<<(NN / 16) / 8, 256, 0, stream>>>(feat, W, outdeg, msg);
    // 4) agg[dst] += msg[src]
    edge_scatter_kernel<<<EE / 256, 256, 0, stream>>>(src, dst, msg, agg);
    // 5) per-graph mean of h = agg*dst_norm + b
    pool_kernel<<<GG, 256, 0, stream>>>(agg, indeg, bvec, pooled);
    // 6) LSTM + MSE loss, one wave32
    lstm_loss_kernel<<<1, 32, 0, stream>>>(pooled, Wih, Whh, bih, bhh, w_r, b_r, target,
                                           (float*)d_out);
}